// LTCEncoder_42159398977654
// MI455X (gfx1250) — compile-verified
//
#include <hip/hip_runtime.h>
#include <cstdint>

#define D_ 768
#define U_ 1536
#define N_ 4096
#define UNFOLDS 6
#define EPS_ 1e-8f
#define L2E 1.4426950408889634f

#define NWG 96
#define TPB 512
#define JB 16
#define STRIPES (TPB / JB)      // 32
#define IPS (U_ / STRIPES)      // 48 recurrent rows per stripe
#define DPS (D_ / STRIPES)      // 24 sensory rows per stripe

typedef float v2f __attribute__((ext_vector_type(2)));
typedef float v8f __attribute__((ext_vector_type(8)));

// LDS byte offset of a generic pointer to __shared__ (LDS aperture keeps the
// AS(3) offset in the low 32 bits of the flat address).
__device__ __forceinline__ unsigned lds_off(const void* p) {
    return (unsigned)(uintptr_t)p;
}

// Async L2->LDS copy of one 32-bit element per lane (GVS addressing mode).
// Tracked by ASYNCcnt; completion via s_wait_asynccnt.
__device__ __forceinline__ void async_ld_b32(unsigned dst_lds, int byte_off,
                                             const void* saddr) {
    asm volatile("global_load_async_to_lds_b32 %0, %1, %2"
                 :: "v"(dst_lds), "v"(byte_off), "s"(saddr) : "memory");
}
__device__ __forceinline__ void wait_async0() {
    asm volatile("s_wait_asynccnt 0x0" ::: "memory");
}

// Pack {a = sigma*log2e, b = a*mu, w, w*erev} so the inner loop is:
//   e = exp2(b - a*v); s = rcp(1+e); den += w*s; num += (w*erev)*s;
__global__ void ltc_pack(const float* __restrict__ sg, const float* __restrict__ mu,
                         const float* __restrict__ w,  const float* __restrict__ er,
                         float4* __restrict__ out, int n, unsigned* __restrict__ bar) {
    int idx = blockIdx.x * blockDim.x + threadIdx.x;
    if (idx == 0 && bar) { bar[0] = 0u; bar[1] = 0u; }   // init grid-barrier state
    if (idx < n) {
        float a = sg[idx] * L2E;
        float4 o;
        o.x = a;
        o.y = a * mu[idx];
        o.z = w[idx];
        o.w = w[idx] * er[idx];
        out[idx] = o;
    }
}

// Sense-reversing grid barrier (persistent-kernel global sync through L2).
// Only wave 0 wrote vbuf, so only it needs the release fence; every wave does
// the acquire fence (per-CU near caches must be invalidated before rereading).
__device__ __forceinline__ void grid_sync(unsigned* cnt, unsigned* gen, int wv) {
    if (wv == 0) __threadfence();     // release vbuf stores
    __syncthreads();
    if (threadIdx.x == 0) {
        unsigned g = atomicAdd(gen, 0u);
        if (atomicAdd(cnt, 1u) == (unsigned)(NWG - 1)) {
            atomicExch(cnt, 0u);
            atomicAdd(gen, 1u);
        } else {
            while (atomicAdd(gen, 0u) == g) { __builtin_amdgcn_s_sleep(2); }
        }
    }
    __syncthreads();
    __threadfence();                  // acquire
}

// Column-sum a STRIPES x 16 tile of partial sums with V_WMMA_F32_16X16X4_F32:
// A = ones(16x4) so D[m,n] += sum_k B[k,n]. One wave, full EXEC (required).
// B layout (4x16 f32, 2 VGPRs): V0 = rows {0,2}, V1 = rows {1,3} split at lane 16.
__device__ __forceinline__ void wmma_colsum(const float (*P)[JB], int lane,
                                            float* __restrict__ outrow) {
    const int n  = lane & 15;
    const int hi = lane >> 4;         // 0: rows k,k+1   1: rows k+2,k+3
    v2f a; a.x = 1.0f; a.y = 1.0f;    // ones A-matrix (16x4)
    v8f acc = {0.f, 0.f, 0.f, 0.f, 0.f, 0.f, 0.f, 0.f};
#pragma unroll
    for (int kb = 0; kb < STRIPES; kb += 4) {
        const int r0 = kb + (hi << 1);
        v2f b; b.x = P[r0][n]; b.y = P[r0 + 1][n];
        acc = __builtin_amdgcn_wmma_f32_16x16x4_f32(
            /*neg_a=*/false, a, /*neg_b=*/false, b,
            /*c_mod=*/(short)0, acc, /*reuse_a=*/false, /*reuse_b=*/false);
    }
    if (lane < 16) outrow[n] = acc[0];   // row M=0 lives in VGPR0, lanes 0-15
}

__global__ __launch_bounds__(TPB, 1)
void ltc_scan(const float* __restrict__ x,    const float* __restrict__ dt,
              const float* __restrict__ iw,   const float* __restrict__ ibv,
              const float* __restrict__ gleak, const float* __restrict__ vleak,
              const float* __restrict__ cm,
              const float4* __restrict__ Wrec, const float4* __restrict__ Wsen,
              float* __restrict__ vbuf, unsigned* __restrict__ bar,
              float* __restrict__ out) {
    __shared__ float v_lds[U_];
    __shared__ float xraw[2][D_];     // async-prefetched raw x rows
    __shared__ float xin_lds[D_];     // affine-mapped input row
    __shared__ float iw_lds[D_];
    __shared__ float ib_lds[D_];
    __shared__ float part_den[STRIPES][JB];
    __shared__ float part_num[STRIPES][JB];
    __shared__ float red_den[JB];
    __shared__ float red_num[JB];
    __shared__ float sden_l[JB];
    __shared__ float snum_l[JB];

    const int tid   = threadIdx.x;
    const int wg    = blockIdx.x;
    const int jbase = wg * JB;
    const int jl    = tid & (JB - 1);
    const int p     = tid >> 4;       // stripe 0..31
    const int lane  = tid & 31;
    const int wv    = tid >> 5;       // wave id (wave32)

    // per-j cell constants, held by the first 16 threads
    float gj = 0.f, vlj = 0.f, cmj = 0.f;
    if (tid < JB) {
        gj  = gleak[jbase + tid];
        vlj = vleak[jbase + tid];
        cmj = cm[jbase + tid];
    }

    for (int i = tid; i < U_; i += TPB) v_lds[i] = 0.f;   // v0 = 0
    for (int d = tid; d < D_; d += TPB) {                 // stage affine params once
        iw_lds[d] = iw[d];
        ib_lds[d] = ibv[d];
    }
    // async prefetch of x row 0 into xraw[0]
    for (int d = tid; d < D_; d += TPB)
        async_ld_b32(lds_off(&xraw[0][d]), d * 4, x);
    __syncthreads();

    int nxt = 0;
    for (int t = 0; t < N_; ++t) {
        // x row t landed in xraw[t&1] (issued at step t-1 / preamble)
        wait_async0();
        __syncthreads();
        const int cb = t & 1;
        for (int d = tid; d < D_; d += TPB)
            xin_lds[d] = xraw[cb][d] * iw_lds[d] + ib_lds[d];
        // overlap: async prefetch of next x row into the other buffer
        if (t + 1 < N_) {
            const float* src = x + (size_t)(t + 1) * D_;
            for (int d = tid; d < D_; d += TPB)
                async_ld_b32(lds_off(&xraw[cb ^ 1][d]), d * 4, src);
        }
        __syncthreads();

        // ---- sensory sums (depend only on x[t]) ----
        {
            float sd = 0.f, sn = 0.f;
            const int dbase = p * DPS;
#pragma unroll 4
            for (int k = 0; k < DPS; ++k) {
                const int d = dbase + k;
                const float4 q = Wsen[(size_t)d * U_ + jbase + jl];
                const float e = __builtin_amdgcn_exp2f(q.y - q.x * xin_lds[d]);
                const float s = __builtin_amdgcn_rcpf(1.0f + e);
                sd += q.z * s;
                sn += q.w * s;
            }
            part_den[p][jl] = sd;
            part_num[p][jl] = sn;
        }
        __syncthreads();
        if (wv == 0)      wmma_colsum(part_den, lane, sden_l);
        else if (wv == 1) wmma_colsum(part_num, lane, snum_l);
        __syncthreads();

        float cmt = 0.f;
        if (tid < JB) {
            float tcl = dt[t];
            tcl = tcl < 0.001f ? 0.001f : tcl;
            cmt = cmj * (float)UNFOLDS / tcl;   // cm / (t / ODE_UNFOLDS)
        }

        // ---- 6 semi-implicit unfolds (globally sequential) ----
        for (int u = 0; u < UNFOLDS; ++u) {
            float dsum = 0.f, nsum = 0.f;
            const int ib0 = p * IPS;
#pragma unroll 4
            for (int k = 0; k < IPS; ++k) {
                const int i = ib0 + k;
                const float4 q = Wrec[(size_t)i * U_ + jbase + jl];
                const float e = __builtin_amdgcn_exp2f(q.y - q.x * v_lds[i]);
                const float s = __builtin_amdgcn_rcpf(1.0f + e);
                dsum += q.z * s;
                nsum += q.w * s;
            }
            part_den[p][jl] = dsum;
            part_num[p][jl] = nsum;
            __syncthreads();
            if (wv == 0)      wmma_colsum(part_den, lane, red_den);
            else if (wv == 1) wmma_colsum(part_num, lane, red_num);
            __syncthreads();

            if (tid < JB) {
                const float vold  = v_lds[jbase + tid];
                const float numer = cmt * vold + gj * vlj + red_num[tid] + snum_l[tid];
                const float denom = cmt + gj + red_den[tid] + sden_l[tid] + EPS_;
                vbuf[nxt * U_ + jbase + tid] = numer / denom;
            }
            grid_sync(bar, bar + 1, wv);

            // async re-broadcast of the full v vector: L2 -> LDS directly
            for (int i = tid; i < U_; i += TPB)
                async_ld_b32(lds_off(&v_lds[i]), (nxt * U_ + i) * 4, vbuf);
            wait_async0();
            nxt ^= 1;
            __syncthreads();
        }
    }

    if (wg == 0) {
        for (int i = tid; i < U_; i += TPB) out[i] = v_lds[i];
    }
}

extern "C" void kernel_launch(void* const* d_in, const int* in_sizes, int n_in,
                              void* d_out, int out_size, void* d_ws, size_t ws_size,
                              hipStream_t stream) {
    const float* x     = (const float*)d_in[0];
    const float* dt    = (const float*)d_in[1];
    const float* iw    = (const float*)d_in[2];
    const float* ibv   = (const float*)d_in[3];
    const float* gleak = (const float*)d_in[4];
    const float* vleak = (const float*)d_in[5];
    const float* cm    = (const float*)d_in[6];
    const float* sigma = (const float*)d_in[7];
    const float* mu    = (const float*)d_in[8];
    const float* w     = (const float*)d_in[9];
    const float* erev  = (const float*)d_in[10];
    const float* ssig  = (const float*)d_in[11];
    const float* smu   = (const float*)d_in[12];
    const float* sw    = (const float*)d_in[13];
    const float* ser   = (const float*)d_in[14];

    char* ws = (char*)d_ws;
    float4* Wrec = (float4*)ws;  ws += (size_t)U_ * U_ * sizeof(float4);  // 37.7 MB
    float4* Wsen = (float4*)ws;  ws += (size_t)D_ * U_ * sizeof(float4);  // 18.9 MB
    float*  vbuf = (float*)ws;   ws += (size_t)2 * U_ * sizeof(float);
    unsigned* bar = (unsigned*)ws;

    const int nrec = U_ * U_;
    const int nsen = D_ * U_;
    ltc_pack<<<(nrec + 255) / 256, 256, 0, stream>>>(sigma, mu, w, erev, Wrec, nrec, bar);
    ltc_pack<<<(nsen + 255) / 256, 256, 0, stream>>>(ssig, smu, sw, ser, Wsen, nsen, nullptr);
    ltc_scan<<<NWG, TPB, 0, stream>>>(x, dt, iw, ibv, gleak, vleak, cm,
                                      Wrec, Wsen, vbuf, bar, (float*)d_out);
}